// STU_14190571946570
// MI455X (gfx1250) — compile-verified
//
#include <hip/hip_runtime.h>

// STU forward for MI455X (gfx1250), fp32 end-to-end using V_WMMA_F32_16X16X4_F32.
//
// Workspace layout (assumes ws_size >= ~61 MB):
//   gpair: K*(L+32) float2   reversed filter PAIRS, 8B-aligned:
//                            gpair[k][p] = (g_k[L-1-p], g_k[L-2-p]), zeros past
//                            the end, where g_k[dt] = eig_vecs[dt,k]*lam_k^0.25.
//                            One aligned b64 load == one A-fragment.
//   Mt   : KY*D*D floats     m_y transposed for the recurrence.
//   phiT : D*(K*D) floats    m_phi transposed: phiT[o][r] = m_phi[r][o].
//   uT   : KU*D*D floats     m_u transposed:  uT[i][o][d] = m_u[o][d][i].
//   xT   : D*L floats        per-batch input transpose: xT[d][s] = x[s][d].
//   conv : L*(K*D) floats    per-batch spectral conv buffer.
//
// deltas are written directly to d_out; the recurrence kernel rewrites d_out in place.

constexpr int Bn  = 4;
constexpr int Ln  = 2048;
constexpr int Dn  = 256;
constexpr int Kn  = 24;
constexpr int KUn = 3;
constexpr int KYn = 2;
constexpr int PADn = 32;              // zero tail on each reversed filter row
constexpr int GLn  = Ln + PADn;       // padded filter stride (in pairs)
constexpr int REDn = Kn * Dn;         // 6144 reduction dim for the m_phi GEMM

typedef __attribute__((ext_vector_type(2))) float v2f;
typedef __attribute__((ext_vector_type(8))) float v8f;

static __device__ __forceinline__ v8f wmma4(v2f a, v2f b, v8f c) {
    // D = A(16x4,f32) * B(4x16,f32) + C(16x16,f32)
    return __builtin_amdgcn_wmma_f32_16x16x4_f32(
        false, a, false, b, (short)0, c, false, false);
}

// ---------------------------------------------------------------------------
// Prep kernels (all tiny vs. the 400 MB total HBM traffic)
// ---------------------------------------------------------------------------
__global__ void stu_prep_gpair(const float* __restrict__ eig_vals,
                               const float* __restrict__ eig_vecs,
                               v2f* __restrict__ gpair) {
    int idx = blockIdx.x * blockDim.x + threadIdx.x;
    if (idx >= Kn * GLn) return;
    int k = idx / GLn;
    int p = idx - k * GLn;
    float w  = powf(eig_vals[k], 0.25f);
    int ta = Ln - 1 - p;              // lag for .x
    int tb = Ln - 2 - p;              // lag for .y
    v2f v;
    v.x = (ta >= 0) ? eig_vecs[ta * Kn + k] * w : 0.0f;
    v.y = (tb >= 0) ? eig_vecs[tb * Kn + k] * w : 0.0f;
    gpair[idx] = v;
}

__global__ void stu_prep_my(const float* __restrict__ m_y,
                            float* __restrict__ Mt) {
    int idx = blockIdx.x * blockDim.x + threadIdx.x;
    if (idx >= KYn * Dn * Dn) return;
    int k   = idx / (Dn * Dn);
    int rem = idx - k * Dn * Dn;
    int j   = rem / Dn;
    int o   = rem - j * Dn;
    Mt[idx] = m_y[(o * KYn + k) * Dn + j];
}

__global__ void stu_prep_phiT(const float* __restrict__ m_phi,
                              float* __restrict__ phiT) {
    int idx = blockIdx.x * blockDim.x + threadIdx.x;
    if (idx >= Dn * REDn) return;
    int o = idx / REDn;
    int r = idx - o * REDn;
    phiT[idx] = m_phi[(size_t)r * Dn + o];
}

__global__ void stu_prep_uT(const float* __restrict__ m_u,
                            float* __restrict__ uT) {
    int idx = blockIdx.x * blockDim.x + threadIdx.x;
    if (idx >= KUn * Dn * Dn) return;
    int i   = idx / (Dn * Dn);
    int rem = idx - i * Dn * Dn;
    int o   = rem / Dn;
    int d   = rem - o * Dn;
    uT[idx] = m_u[((size_t)o * Dn + d) * KUn + i];
}

__global__ void stu_prep_xT(const float* __restrict__ x,   // (L, D) one batch
                            float* __restrict__ xT) {      // (D, L)
    int idx = blockIdx.x * blockDim.x + threadIdx.x;
    if (idx >= Dn * Ln) return;
    int d = idx / Ln;
    int s = idx - d * Ln;
    xT[idx] = x[(size_t)s * Dn + d];
}

// ---------------------------------------------------------------------------
// Stage 1: causal Toeplitz conv. One wave computes a 32(t) x 64(d) block
// (2x4 register tiles) for one k: 6 b64 loads feed 8 WMMAs per K-step
// (0.75 loads/WMMA). Branch-free; reversed/padded filter absorbs causality.
//   conv[t][k][d] = sum_{s<=t} g[k][t-s] * x[s][d]
// grid: (L/32, D/64, K), block: 32 (one wave, EXEC all ones)
// ---------------------------------------------------------------------------
__global__ __launch_bounds__(32) void stu_conv_kernel(
    const float* __restrict__ xT,     // (D, L), one batch, transposed
    const v2f* __restrict__ gpair,    // (K, L+PAD) reversed filter pairs
    float* __restrict__ conv) {       // (L, K*D)
    const int lane = threadIdx.x;
    const int row  = lane & 15;       // A-row / B-col within a 16x16 tile
    const int hi   = lane >> 4;       // selects K-pair {2*hi, 2*hi+1}
    const int t0   = blockIdx.x * 32;
    const int d0   = blockIdx.y * 64;
    const int k    = blockIdx.z;

    // gl[s0] = gpair[k][(Ln-1) - (t0+row) + s0 + 2*hi] = (g[t-sa], g[t-sa-1])
    const v2f* gl = gpair + (size_t)k * GLn + ((Ln - 1) - (t0 + row)) + 2 * hi;
    const float* xc0 = xT + (size_t)(d0 + row) * Ln + 2 * hi;
    const float* xc1 = xT + (size_t)(d0 + 16 + row) * Ln + 2 * hi;
    const float* xc2 = xT + (size_t)(d0 + 32 + row) * Ln + 2 * hi;
    const float* xc3 = xT + (size_t)(d0 + 48 + row) * Ln + 2 * hi;

    v8f cA0 = {}, cA1 = {}, cA2 = {}, cA3 = {};   // t-rows t0..t0+15
    v8f cB0 = {}, cB1 = {}, cB2 = {}, cB3 = {};   // t-rows t0+16..t0+31

    const int smax = t0 + 32;         // pad handles s > t
#pragma unroll 4
    for (int s0 = 0; s0 < smax; s0 += 4) {
        v2f a0 = gl[s0];              // (g[u], g[u-1])
        v2f a1 = gl[s0 - 16];         // (g[u+16], g[u+15])
        v2f b0 = *(const v2f*)(xc0 + s0);   // (x[sa][c], x[sa+1][c]) aligned
        v2f b1 = *(const v2f*)(xc1 + s0);
        v2f b2 = *(const v2f*)(xc2 + s0);
        v2f b3 = *(const v2f*)(xc3 + s0);
        cA0 = wmma4(a0, b0, cA0);
        cA1 = wmma4(a0, b1, cA1);
        cA2 = wmma4(a0, b2, cA2);
        cA3 = wmma4(a0, b3, cA3);
        cB0 = wmma4(a1, b0, cB0);
        cB1 = wmma4(a1, b1, cB1);
        cB2 = wmma4(a1, b2, cB2);
        cB3 = wmma4(a1, b3, cB3);
    }

#pragma unroll
    for (int r = 0; r < 8; ++r) {
        const int m = r + 8 * hi;     // C/D layout: vgpr = M%8, lane half = M/8
        float* p0 = conv + (size_t)(t0 + m) * REDn + k * Dn + d0 + row;
        float* p1 = conv + (size_t)(t0 + 16 + m) * REDn + k * Dn + d0 + row;
        p0[0]  = cA0[r];
        p0[16] = cA1[r];
        p0[32] = cA2[r];
        p0[48] = cA3[r];
        p1[0]  = cB0[r];
        p1[16] = cB1[r];
        p1[32] = cB2[r];
        p1[48] = cB3[r];
    }
}

// ---------------------------------------------------------------------------
// Stage 2: deltas = conv(L x 6144) @ m_phi + AR(shifted m_u GEMMs).
// One wave computes a 32(t) x 32(o) block (kept 2x2 for grid parallelism:
// this stage has 4x fewer FLOPs than stage 1). All fragments are aligned
// b64 loads.
// grid: (L/32, D/32), block: 32
// ---------------------------------------------------------------------------
__global__ __launch_bounds__(32) void stu_delta_kernel(
    const float* __restrict__ conv,   // (L, K*D)
    const float* __restrict__ phiT,   // (D, K*D): phiT[o][r] = m_phi[r][o]
    const float* __restrict__ x,      // (L, D), same batch (row-major)
    const float* __restrict__ uT,     // (KU, D, D): uT[i][o][d] = m_u[o][d][i]
    float* __restrict__ out) {        // (L, D) deltas for this batch
    const int lane = threadIdx.x;
    const int row  = lane & 15;
    const int hi   = lane >> 4;
    const int t0   = blockIdx.x * 32;
    const int o0   = blockIdx.y * 32;

    v8f c00 = {}, c01 = {}, c10 = {}, c11 = {};

    // Main spectral GEMM: contiguous 6144-long reduction, all b64 fragments.
    const float* ar0 = conv + (size_t)(t0 + row) * REDn + 2 * hi;
    const float* ar1 = conv + (size_t)(t0 + 16 + row) * REDn + 2 * hi;
    const float* bp0 = phiT + (size_t)(o0 + row) * REDn + 2 * hi;
    const float* bp1 = phiT + (size_t)(o0 + 16 + row) * REDn + 2 * hi;
#pragma unroll 4
    for (int r0 = 0; r0 < REDn; r0 += 4) {
        v2f a0 = *(const v2f*)(ar0 + r0);
        v2f a1 = *(const v2f*)(ar1 + r0);
        v2f b0 = *(const v2f*)(bp0 + r0);
        v2f b1 = *(const v2f*)(bp1 + r0);
        c00 = wmma4(a0, b0, c00);
        c01 = wmma4(a0, b1, c01);
        c10 = wmma4(a1, b0, c10);
        c11 = wmma4(a1, b1, c11);
    }

    // AR term: sum_i m_u[:,:,i] @ x[t-i]. Only rows t<KU can underflow;
    // clamp the index and scale by a 0/1 mask (branch-free).
    for (int i = 0; i < KUn; ++i) {
        const int ts0 = t0 + row - i;
        const float msk = (ts0 >= 0) ? 1.0f : 0.0f;
        const int tc0 = (ts0 >= 0) ? ts0 : 0;
        const int ts1 = t0 + 16 + row - i;          // always >= 14 >= 0
        const float* xr0 = x + (size_t)tc0 * Dn + 2 * hi;
        const float* xr1 = x + (size_t)ts1 * Dn + 2 * hi;
        const float* bu0 = uT + ((size_t)i * Dn + o0 + row) * Dn + 2 * hi;
        const float* bu1 = uT + ((size_t)i * Dn + o0 + 16 + row) * Dn + 2 * hi;
#pragma unroll 4
        for (int d = 0; d < Dn; d += 4) {
            v2f a0 = *(const v2f*)(xr0 + d);
            a0.x *= msk;
            a0.y *= msk;
            v2f a1 = *(const v2f*)(xr1 + d);
            v2f b0 = *(const v2f*)(bu0 + d);
            v2f b1 = *(const v2f*)(bu1 + d);
            c00 = wmma4(a0, b0, c00);
            c01 = wmma4(a0, b1, c01);
            c10 = wmma4(a1, b0, c10);
            c11 = wmma4(a1, b1, c11);
        }
    }

#pragma unroll
    for (int r = 0; r < 8; ++r) {
        const int m = r + 8 * hi;
        float* p0 = out + (size_t)(t0 + m) * Dn + o0 + row;
        float* p1 = out + (size_t)(t0 + 16 + m) * Dn + o0 + row;
        p0[0]  = c00[r];
        p0[16] = c01[r];
        p1[0]  = c10[r];
        p1[16] = c11[r];
    }
}

// ---------------------------------------------------------------------------
// Stage 3: y_t = delta_t + M0 y_{t-1} + M1 y_{t-2}, in place over d_out.
// One workgroup per batch; D threads; state double-buffered in LDS;
// Mt reads are coalesced and L2-resident (512 KB).
// ---------------------------------------------------------------------------
__global__ __launch_bounds__(Dn) void stu_recur_kernel(
    const float* __restrict__ Mt,    // (KY, D, D): Mt[k][j][o]
    float* __restrict__ y) {         // (B, L, D): in=deltas, out=y
    const int b = blockIdx.x;
    const int o = threadIdx.x;
    __shared__ float y1[Dn];
    __shared__ float y2[Dn];
    y1[o] = 0.0f;
    y2[o] = 0.0f;
    __syncthreads();

    float* yb = y + (size_t)b * Ln * Dn;
    const float* M0 = Mt + o;
    const float* M1 = Mt + Dn * Dn + o;

    for (int t = 0; t < Ln; ++t) {
        float acc = yb[(size_t)t * Dn + o];
        float acc2 = 0.0f;
#pragma unroll 8
        for (int j = 0; j < Dn; ++j) {
            acc  += M0[j * Dn] * y1[j];
            acc2 += M1[j * Dn] * y2[j];
        }
        acc += acc2;
        __syncthreads();             // everyone done reading y1/y2
        yb[(size_t)t * Dn + o] = acc;
        y2[o] = y1[o];
        y1[o] = acc;
        __syncthreads();             // state visible for next step
    }
}

// ---------------------------------------------------------------------------
extern "C" void kernel_launch(void* const* d_in, const int* in_sizes, int n_in,
                              void* d_out, int out_size, void* d_ws, size_t ws_size,
                              hipStream_t stream) {
    const float* x     = (const float*)d_in[0];   // (B, L, D)
    const float* m_u   = (const float*)d_in[1];   // (D, D, KU)
    const float* m_phi = (const float*)d_in[2];   // (K*D, D)
    const float* m_y   = (const float*)d_in[3];   // (D, KY, D)
    const float* evals = (const float*)d_in[4];   // (K,)
    const float* evecs = (const float*)d_in[5];   // (L, K)
    float* out = (float*)d_out;                   // (B, L, D)

    char* ws = (char*)d_ws;
    size_t off = 0;
    v2f*   gpair = (v2f*)(ws + off);   off += (size_t)Kn * GLn * 8;
    float* Mt    = (float*)(ws + off); off += (size_t)KYn * Dn * Dn * 4;
    float* phiT  = (float*)(ws + off); off += (size_t)Dn * REDn * 4;
    float* uT    = (float*)(ws + off); off += (size_t)KUn * Dn * Dn * 4;
    float* xT    = (float*)(ws + off); off += (size_t)Dn * Ln * 4;
    float* conv  = (float*)(ws + off);

    stu_prep_gpair<<<(Kn * GLn + 255) / 256, 256, 0, stream>>>(evals, evecs, gpair);
    stu_prep_my<<<(KYn * Dn * Dn + 255) / 256, 256, 0, stream>>>(m_y, Mt);
    stu_prep_phiT<<<(Dn * REDn + 255) / 256, 256, 0, stream>>>(m_phi, phiT);
    stu_prep_uT<<<(KUn * Dn * Dn + 255) / 256, 256, 0, stream>>>(m_u, uT);

    for (int b = 0; b < Bn; ++b) {
        const float* xb = x + (size_t)b * Ln * Dn;
        float* ob = out + (size_t)b * Ln * Dn;
        stu_prep_xT<<<(Dn * Ln + 255) / 256, 256, 0, stream>>>(xb, xT);
        stu_conv_kernel<<<dim3(Ln / 32, Dn / 64, Kn), 32, 0, stream>>>(xT, gpair, conv);
        stu_delta_kernel<<<dim3(Ln / 32, Dn / 32), 32, 0, stream>>>(conv, phiT, xb, uT, ob);
    }

    stu_recur_kernel<<<Bn, Dn, 0, stream>>>(Mt, out);
}